// DeepseekV4HCACompressor_13761075216598
// MI455X (gfx1250) — compile-verified
//
#include <hip/hip_runtime.h>
#include <cstdint>

// DeepseekV4 HCA compressor for MI455X (gfx1250), fp32 WMMA path.
// hA streamed via async global->LDS b128 (ASYNCcnt double buffer);
// weights pipelined through VGPRs into a pre-paired float2 LDS layout
// so each WMMA B-fragment is a single ds_load_b64 (no repack movs).
// B=2, S=8192, H=2048, m=128, d=128, T=64.

#define HDIM   2048
#define HD     128      // HEAD_DIM
#define MCOMP  128      // COMPRESS_RATE
#define TBLK   64       // S / MCOMP
#define NBATCH 2
#define BK     16       // K-chunk staged per buffer
#define NKB    (HDIM / BK)          // 128 chunks
#define BUFSZ  6144                 // floats: 2048 hA + 2048 wK2 + 2048 wG2

typedef float v2f __attribute__((ext_vector_type(2)));
typedef float v8f __attribute__((ext_vector_type(8)));
typedef int   vi4 __attribute__((vector_size(16)));   // builtin param type

#if defined(__AMDGCN__) && \
    __has_builtin(__builtin_amdgcn_global_load_async_to_lds_b128) && \
    __has_builtin(__builtin_amdgcn_s_wait_asynccnt)
#define USE_ASYNC 1
#else
#define USE_ASYNC 0
#endif

#if USE_ASYNC
__device__ __forceinline__ void async_cp16(const float* g, float* l)
{
    __builtin_amdgcn_global_load_async_to_lds_b128(
        (__attribute__((address_space(1))) vi4*)(uintptr_t)g,
        (__attribute__((address_space(3))) vi4*)(uintptr_t)l,
        0, 0);
}
#endif

// hA chunk: 128 rows x 16 k, row-major; 512 b128 chunks / 256 threads.
__device__ __forceinline__ void stage_hA(const float* __restrict__ hblk,
                                         float* __restrict__ buf,
                                         int k0, int tid)
{
    #pragma unroll
    for (int i = 0; i < 2; ++i) {
        int idx = tid + i * 256;
        int row = idx >> 2, cv = idx & 3;        // 4 chunks per 16-float row
#if USE_ASYNC
        async_cp16(hblk + (size_t)row * HDIM + k0 + cv * 4,
                   buf + row * BK + cv * 4);
#else
        *(float4*)(buf + row * BK + cv * 4) =
            *(const float4*)(hblk + (size_t)row * HDIM + k0 + cv * 4);
#endif
    }
}

// Weights for one chunk held in VGPRs: 4 float2 per matrix per thread.
struct WRegs { float2 k[4]; float2 g[4]; };

__device__ __forceinline__ void wload(WRegs& r,
                                      const float* __restrict__ w_kv,
                                      const float* __restrict__ w_gate,
                                      int k0, int tid)
{
    #pragma unroll
    for (int i = 0; i < 4; ++i) {
        int idx = tid + i * 256;                 // 0..1023: p = k-pair, n = col
        int p = idx >> 7, n = idx & 127;
        int kk = k0 + 2 * p;
        r.k[i].x = w_kv[(size_t)kk * HD + n];
        r.k[i].y = w_kv[(size_t)(kk + 1) * HD + n];
        r.g[i].x = w_gate[(size_t)kk * HD + n];
        r.g[i].y = w_gate[(size_t)(kk + 1) * HD + n];
    }
}

__device__ __forceinline__ void wstore(float* __restrict__ buf,
                                       const WRegs& r, int tid)
{
    #pragma unroll
    for (int i = 0; i < 4; ++i) {
        int idx = tid + i * 256;
        *(float2*)(buf + 2048 + idx * 2) = r.k[i];   // wK2[p][n] pair
        *(float2*)(buf + 4096 + idx * 2) = r.g[i];   // wG2[p][n] pair
    }
}

// One K-chunk of WMMA: 4 k-steps x (8 kv + 8 gate) tiles.
__device__ __forceinline__ void gemm_chunk(const float* __restrict__ buf,
                                           int wave, int lmod, int lhalf,
                                           v8f acc_k[8], v8f acc_g[8])
{
    const float* hA  = buf;
    const float* wK2 = buf + 2048;
    const float* wG2 = buf + 4096;
    #pragma unroll
    for (int ks = 0; ks < 4; ++ks) {
        // A 16x4 f32: lane holds row (16*wave + lmod), k = ks*4 + 2*lhalf + {0,1}
        v2f a = *(const v2f*)(hA + (16 * wave + lmod) * BK + ks * 4 + 2 * lhalf);
        const int p = ks * 2 + lhalf;            // k-pair index 0..7
        #pragma unroll
        for (int ct = 0; ct < 8; ++ct) {
            const int n = ct * 16 + lmod;
            v2f bk = *(const v2f*)(wK2 + (p * 128 + n) * 2);
            v2f bg = *(const v2f*)(wG2 + (p * 128 + n) * 2);
            acc_k[ct] = __builtin_amdgcn_wmma_f32_16x16x4_f32(
                false, a, false, bk, (short)0, acc_k[ct], false, false);
            acc_g[ct] = __builtin_amdgcn_wmma_f32_16x16x4_f32(
                false, a, false, bg, (short)0, acc_g[ct], false, false);
        }
    }
}

// ---------------------------------------------------------------------------
// Kernel 1: per (b,t) block of 128 rows:
//   kv = h @ w_kv, gate = h @ w_gate  (128x2048 @ 2048x128, fp32 WMMA)
//   w  = softmax(gate + position_bias, axis=m);  comp = sum_m w*kv
// ---------------------------------------------------------------------------
__global__ __launch_bounds__(256)
void hca_block_kernel(const float* __restrict__ hidden,
                      const float* __restrict__ w_kv,
                      const float* __restrict__ w_gate,
                      const float* __restrict__ pos_bias,
                      float* __restrict__ comp)
{
    __shared__ float smem[2 * BUFSZ];        // 48 KB, double buffered

    const int tid   = threadIdx.x;
    const int wave  = tid >> 5;              // 0..7 -> 16-row stripe
    const int lane  = tid & 31;
    const int lmod  = lane & 15;
    const int lhalf = lane >> 4;

    const int blk = blockIdx.x;              // b*TBLK + t
    const float* hblk = hidden + (size_t)blk * MCOMP * HDIM;

    v8f acc_k[8] = {};
    v8f acc_g[8] = {};

    // Prologue: fully stage chunk 0.
    WRegs wr;
    stage_hA(hblk, smem, 0, tid);
    wload(wr, w_kv, w_gate, 0, tid);
    wstore(smem, wr, tid);

    for (int kb = 0; kb < NKB; ++kb) {
        float* cur = smem + (kb & 1) * BUFSZ;
        float* nxt = smem + ((kb + 1) & 1) * BUFSZ;
        if (kb + 1 < NKB) {
            stage_hA(hblk, nxt, (kb + 1) * BK, tid);     // async, overlaps
            wload(wr, w_kv, w_gate, (kb + 1) * BK, tid); // global->VGPR, overlaps
#if USE_ASYNC
            __builtin_amdgcn_s_wait_asynccnt(2);         // cur hA's 2 done
#endif
        } else {
#if USE_ASYNC
            __builtin_amdgcn_s_wait_asynccnt(0);
#endif
        }
        __syncthreads();
        gemm_chunk(cur, wave, lmod, lhalf, acc_k, acc_g);
        if (kb + 1 < NKB) wstore(nxt, wr, tid);          // after compute
        __syncthreads();
    }

    // --- fold position_bias into gate accumulators.
    // C/D layout: lane holds rows m_local = v + 8*lhalf (v = VGPR idx 0..7).
    #pragma unroll
    for (int ct = 0; ct < 8; ++ct) {
        int n = ct * 16 + lmod;
        #pragma unroll
        for (int v = 0; v < 8; ++v) {
            int m = 16 * wave + 8 * lhalf + v;
            acc_g[ct][v] += pos_bias[m * HD + n];
        }
    }

    // --- softmax over m (128 rows) per column via 16-slot LDS reduction.
    const int slot = wave * 2 + lhalf;       // 0..15, each owns 8 rows/column
    float* rmax  = smem;                     // 16 x 128
    float* rsum  = smem + 2048;              // 16 x 128
    float* rwsum = smem + 4096;              // 16 x 128
    float* cmax  = smem + 6144;              // 128

    #pragma unroll
    for (int ct = 0; ct < 8; ++ct) {
        int n = ct * 16 + lmod;
        float pm = -3.402823466e38f;
        #pragma unroll
        for (int v = 0; v < 8; ++v) pm = fmaxf(pm, acc_g[ct][v]);
        rmax[slot * 128 + n] = pm;
    }
    __syncthreads();
    if (tid < 128) {
        float m = rmax[tid];
        #pragma unroll
        for (int s = 1; s < 16; ++s) m = fmaxf(m, rmax[s * 128 + tid]);
        cmax[tid] = m;
    }
    __syncthreads();
    #pragma unroll
    for (int ct = 0; ct < 8; ++ct) {
        int n = ct * 16 + lmod;
        float cm = cmax[n];
        float ps = 0.f, pw = 0.f;
        #pragma unroll
        for (int v = 0; v < 8; ++v) {
            float e = __expf(acc_g[ct][v] - cm);
            ps += e;
            pw += e * acc_k[ct][v];
        }
        rsum[slot * 128 + n]  = ps;
        rwsum[slot * 128 + n] = pw;
    }
    __syncthreads();
    if (tid < 128) {
        float s = 0.f, w = 0.f;
        #pragma unroll
        for (int q = 0; q < 16; ++q) {
            s += rsum[q * 128 + tid];
            w += rwsum[q * 128 + tid];
        }
        comp[(size_t)blk * HD + tid] = w / s;
    }
}

// ---------------------------------------------------------------------------
// Kernel 2: RMS-norm over d, then RoPE on last 64 dims
// ---------------------------------------------------------------------------
__global__ __launch_bounds__(128)
void hca_norm_rope_kernel(const float* __restrict__ comp,
                          const float* __restrict__ kvw,
                          float* __restrict__ out)
{
    __shared__ float red[128];
    __shared__ float nv[128];
    const int tid = threadIdx.x;
    const int row = blockIdx.x;              // b*TBLK + t
    const int t   = row & (TBLK - 1);

    float c = comp[(size_t)row * HD + tid];
    red[tid] = c * c;
    __syncthreads();
    #pragma unroll
    for (int s = 64; s > 0; s >>= 1) {
        if (tid < s) red[tid] += red[tid + s];
        __syncthreads();
    }
    float scale = rsqrtf(red[0] * (1.0f / HD) + 1e-6f);
    float val = c * scale * kvw[tid];
    nv[tid] = val;
    __syncthreads();

    float o = val;
    if (tid >= 64) {
        int j = tid - 64;                    // rope-slice index
        int i = j >> 1;                      // repeat-2 frequency index
        float inv = __powf(10000.0f, -(float)i * (1.0f / 32.0f));
        float f = (float)(t * MCOMP) * inv;
        float cv = __cosf(f), sv = __sinf(f);
        float rot = (j < 32) ? -nv[64 + j + 32] : nv[64 + j - 32];
        o = val * cv + rot * sv;
    }
    out[(size_t)row * HD + tid] = o;
}

// ---------------------------------------------------------------------------
extern "C" void kernel_launch(void* const* d_in, const int* in_sizes, int n_in,
                              void* d_out, int out_size, void* d_ws, size_t ws_size,
                              hipStream_t stream)
{
    const float* hidden = (const float*)d_in[0];   // (2, 8192, 2048) f32
    // d_in[1] q_residual, d_in[2] position_ids: unused by the reference math
    const float* w_kv   = (const float*)d_in[3];   // (2048, 128)
    const float* w_gate = (const float*)d_in[4];   // (2048, 128)
    const float* p_bias = (const float*)d_in[5];   // (128, 128)
    const float* kvw    = (const float*)d_in[6];   // (128,)

    float* comp = (float*)d_ws;                    // (128, 128) = 64 KB scratch
    float* out  = (float*)d_out;                   // (2, 1, 64, 128)

    hca_block_kernel<<<dim3(NBATCH * TBLK), dim3(256), 0, stream>>>(
        hidden, w_kv, w_gate, p_bias, comp);
    hca_norm_rope_kernel<<<dim3(NBATCH * TBLK), dim3(128), 0, stream>>>(
        comp, kvw, out);
}